// P2A_54296976556427
// MI455X (gfx1250) — compile-verified
//
#include <hip/hip_runtime.h>
#include <math.h>

// ---------------------------------------------------------------------------
// CDNA5 (gfx1250) implementation of the 2-block TransformerConv GNN.
// Dense GEMMs run on the fp32 WMMA pipe (v_wmma_f32_16x16x4_f32), register-
// blocked 4 M-tiles per wave so each B (weight) fragment feeds 4 WMMAs.
// Edge softmax/aggregation use native f32 atomics. Working set (~100MB) is
// L2-resident on MI455X (192MB L2).
// ---------------------------------------------------------------------------

typedef float v2f __attribute__((ext_vector_type(2)));
typedef float v8f __attribute__((ext_vector_type(8)));

#define N_NODES_C  50000
#define N_ACTORS_C 1024
#define NE_EDGES_C 800000
#define NA_EDGES_C 50000
#define DHID       64      // feature dim (both node and actor)
#define HC_C       128     // H*C = 2*64
#define HEADS      2
#define CHEAD      64

// ======================= fp32 WMMA GEMM + bias (+ReLU) ======================
// C[M,N] = act(A[M,K] @ B[K,N] + bias[N]).
// One wave computes a 64x16 output slab (4 stacked 16x16 tiles) so a single
// B fragment is reused by 4 WMMAs per k-step.
// Requires M%16==0, N%16==0, K%4==0 (always true here).
// A-frag (16x4 f32): lane L: m=L&15, half=L>>4; a[i] = A[m][k0+2*half+i]
// B-frag (4x16 f32): lane L: n=L&15;            b[i] = B[k0+2*half+i][n]
// D (16x16 f32):     acc[j] -> C[tm*16 + j + 8*half][n]
template <bool RELU>
__global__ void gemm_bias_wmma(const float* __restrict__ A,
                               const float* __restrict__ B,
                               const float* __restrict__ bias,
                               float* __restrict__ C,
                               int M, int K, int N) {
  const int waveId = threadIdx.x >> 5;
  const int lane   = threadIdx.x & 31;
  const int tilesN = N >> 4;
  const int tilesM = M >> 4;
  const int mGroups = (tilesM + 3) >> 2;
  const int group = blockIdx.x * (blockDim.x >> 5) + waveId;  // wave-uniform
  if (group >= mGroups * tilesN) return;                      // uniform exit

  const int gm = group / tilesN;
  const int tn = group - gm * tilesN;
  const int half = lane >> 4;
  const int l16  = lane & 15;
  const int colB = tn * 16 + l16;

  // Row base per sub-tile; tail tiles clamp to the last valid tile (their
  // stores are skipped) so loads stay in-bounds and EXEC stays all-ones.
  int rowBase[4];
#pragma unroll
  for (int t = 0; t < 4; ++t) {
    int tm = gm * 4 + t;
    if (tm > tilesM - 1) tm = tilesM - 1;
    rowBase[t] = tm * 16;
  }

  v8f acc[4] = {};
  for (int k0 = 0; k0 < K; k0 += 4) {
    const int ka = k0 + 2 * half;
    v2f b;
    b.x = B[(size_t)ka * N + colB];
    b.y = B[(size_t)(ka + 1) * N + colB];
#pragma unroll
    for (int t = 0; t < 4; ++t) {
      v2f a = *(const v2f*)(A + (size_t)(rowBase[t] + l16) * K + ka);
      acc[t] = __builtin_amdgcn_wmma_f32_16x16x4_f32(false, a, false, b,
                                                     (short)0, acc[t],
                                                     false, false);
    }
  }

  const float bc = bias[colB];
#pragma unroll
  for (int t = 0; t < 4; ++t) {
    if (gm * 4 + t >= tilesM) break;  // wave-uniform
#pragma unroll
    for (int j = 0; j < 8; ++j) {
      float val = acc[t][j] + bc;
      if (RELU) val = fmaxf(val, 0.0f);
      C[(size_t)(rowBase[t] + j + 8 * half) * N + colB] = val;
    }
  }
}

// ============================ small utilities ===============================
__global__ void fill_kernel(float* __restrict__ p, float v, int n) {
  int i = blockIdx.x * blockDim.x + threadIdx.x;
  if (i < n) p[i] = v;
}

__device__ __forceinline__ void atomicMaxF(float* addr, float val) {
  // bit-trick float max: valid with -inf initialization, no NaNs expected
  if (val >= 0.0f) atomicMax((int*)addr, __float_as_int(val));
  else             atomicMin((unsigned int*)addr, __float_as_uint(val));
}

// =========================== edge-attention kernels =========================
// alpha[e,h] = <q[dst[e],h,:], k[src[e],h,:]> / 8 ; running segment max
__global__ void edge_alpha_kernel(const float* __restrict__ q,
                                  const float* __restrict__ k,
                                  const int* __restrict__ src,
                                  const int* __restrict__ dst,
                                  float* __restrict__ alpha,
                                  float* __restrict__ amax,
                                  int E) {
  int idx = blockIdx.x * blockDim.x + threadIdx.x;
  if (idx >= E * HEADS) return;
  int e = idx >> 1, h = idx & 1;
  int s = src[e], d = dst[e];
  const float* __restrict__ qp = q + (size_t)d * HC_C + h * CHEAD;
  const float* __restrict__ kp = k + (size_t)s * HC_C + h * CHEAD;
  float acc = 0.0f;
#pragma unroll 4
  for (int c = 0; c < CHEAD; c += 4) {
    float4 qv = *(const float4*)(qp + c);
    float4 kv = *(const float4*)(kp + c);
    acc += qv.x * kv.x + qv.y * kv.y + qv.z * kv.z + qv.w * kv.w;
  }
  acc *= 0.125f;  // 1/sqrt(64)
  alpha[idx] = acc;
  atomicMaxF(&amax[(size_t)d * HEADS + h], acc);
}

// ex = exp(alpha - amax[dst]); den[dst] += ex
__global__ void edge_exp_kernel(const int* __restrict__ dst,
                                float* __restrict__ alpha,
                                const float* __restrict__ amax,
                                float* __restrict__ den,
                                int E) {
  int idx = blockIdx.x * blockDim.x + threadIdx.x;
  if (idx >= E * HEADS) return;
  int e = idx >> 1, h = idx & 1;
  int d = dst[e];
  float x = expf(alpha[idx] - amax[(size_t)d * HEADS + h]);
  alpha[idx] = x;
  atomicAdd(&den[(size_t)d * HEADS + h], x);
}

// agg[dst, i] += v[src, i] * ex[e,h]/den[dst,h]   (i in [0,128), h = i/64)
__global__ void edge_agg_kernel(const int* __restrict__ src,
                                const int* __restrict__ dst,
                                const float* __restrict__ v,
                                const float* __restrict__ ex,
                                const float* __restrict__ den,
                                float* __restrict__ agg,
                                int E) {
  int idx = blockIdx.x * blockDim.x + threadIdx.x;
  if (idx >= E * HC_C) return;
  int e = idx >> 7, i = idx & 127;
  int h = i >> 6;
  int s = src[e], d = dst[e];
  float w = ex[(size_t)e * HEADS + h] / den[(size_t)d * HEADS + h];
  atomicAdd(&agg[(size_t)d * HC_C + i], v[(size_t)s * HC_C + i] * w);
}

// ================== fused residual + LayerNorm (D = 64) =====================
// out[row,:] = LN(a[row,:] + r[row,:]) * g + b     one wave per row
__global__ void ln_residual_kernel(const float* __restrict__ a,
                                   const float* __restrict__ r,
                                   const float* __restrict__ g,
                                   const float* __restrict__ b,
                                   float* __restrict__ out, int n) {
  int wid = threadIdx.x >> 5, lane = threadIdx.x & 31;
  int row = blockIdx.x * (blockDim.x >> 5) + wid;
  if (row >= n) return;
  const float* ap = a + (size_t)row * DHID;
  const float* rp = r + (size_t)row * DHID;
  float x0 = ap[lane] + rp[lane];
  float x1 = ap[lane + 32] + rp[lane + 32];
  float s = x0 + x1;
  for (int m = 16; m >= 1; m >>= 1) s += __shfl_xor(s, m, 32);
  float mean = s * (1.0f / 64.0f);
  float d0 = x0 - mean, d1 = x1 - mean;
  float vs = d0 * d0 + d1 * d1;
  for (int m = 16; m >= 1; m >>= 1) vs += __shfl_xor(vs, m, 32);
  float rstd = rsqrtf(vs * (1.0f / 64.0f) + 1e-5f);
  float* op = out + (size_t)row * DHID;
  op[lane]      = d0 * rstd * g[lane]      + b[lane];
  op[lane + 32] = d1 * rstd * g[lane + 32] + b[lane + 32];
}

// ============================ host orchestration ============================
struct LayerP {
  const float *Wq, *bq, *Wk, *bk, *Wv, *bv, *Ws, *bs;
  const float *g1, *b1n, *g2, *b2n, *W1, *b1f, *W2, *b2f;
};

static LayerP layer_params(const float* const* P, int l) {
  LayerP p;
  p.Wq  = P[0]  + (size_t)l * DHID * HC_C;  p.bq  = P[1]  + (size_t)l * HC_C;
  p.Wk  = P[2]  + (size_t)l * DHID * HC_C;  p.bk  = P[3]  + (size_t)l * HC_C;
  p.Wv  = P[4]  + (size_t)l * DHID * HC_C;  p.bv  = P[5]  + (size_t)l * HC_C;
  p.Ws  = P[6]  + (size_t)l * HC_C * DHID;  p.bs  = P[7]  + (size_t)l * DHID;
  p.g1  = P[8]  + (size_t)l * DHID;         p.b1n = P[9]  + (size_t)l * DHID;
  p.g2  = P[10] + (size_t)l * DHID;         p.b2n = P[11] + (size_t)l * DHID;
  p.W1  = P[12] + (size_t)l * DHID * HC_C;  p.b1f = P[13] + (size_t)l * HC_C;
  p.W2  = P[14] + (size_t)l * HC_C * DHID;  p.b2f = P[15] + (size_t)l * DHID;
  return p;
}

static inline dim3 gemm_grid(int M, int N) {
  int mGroups = ((M >> 4) + 3) >> 2;          // 4 M-tiles per wave
  int waves = mGroups * (N >> 4);
  return dim3((waves + 3) / 4);               // 4 waves (128 threads) / block
}

static void run_block(const float* x_src, int n_src,
                      const float* x_dst, int n_dst,
                      const int* src, const int* dst, int E,
                      const LayerP& p, float* out,
                      float* qbuf, float* kbuf, float* vbuf,
                      float* exb, float* amax, float* den,
                      hipStream_t stream) {
  // q/k/v projections (WMMA fp32)
  gemm_bias_wmma<false><<<gemm_grid(n_dst, HC_C), 128, 0, stream>>>(
      x_dst, p.Wq, p.bq, qbuf, n_dst, DHID, HC_C);
  gemm_bias_wmma<false><<<gemm_grid(n_src, HC_C), 128, 0, stream>>>(
      x_src, p.Wk, p.bk, kbuf, n_src, DHID, HC_C);
  gemm_bias_wmma<false><<<gemm_grid(n_src, HC_C), 128, 0, stream>>>(
      x_src, p.Wv, p.bv, vbuf, n_src, DHID, HC_C);

  // scatter softmax over edges
  const int nd2 = n_dst * HEADS;
  fill_kernel<<<(nd2 + 255) / 256, 256, 0, stream>>>(amax, -INFINITY, nd2);
  fill_kernel<<<(nd2 + 255) / 256, 256, 0, stream>>>(den, 0.0f, nd2);
  const int EH = E * HEADS;
  edge_alpha_kernel<<<(EH + 255) / 256, 256, 0, stream>>>(qbuf, kbuf, src, dst,
                                                          exb, amax, E);
  edge_exp_kernel<<<(EH + 255) / 256, 256, 0, stream>>>(dst, exb, amax, den, E);

  // weighted scatter-add aggregation (agg reuses qbuf; q is dead now)
  float* agg = qbuf;
  const int aggN = n_dst * HC_C;
  fill_kernel<<<(aggN + 255) / 256, 256, 0, stream>>>(agg, 0.0f, aggN);
  const int EI = E * HC_C;
  edge_agg_kernel<<<(EI + 255) / 256, 256, 0, stream>>>(src, dst, vbuf, exb,
                                                        den, agg, E);

  // output projection + LN1 + FFN + LN2 (kbuf reused as two 64-wide temps)
  float* t64a = kbuf;
  float* t64b = kbuf + (size_t)n_dst * DHID;
  gemm_bias_wmma<false><<<gemm_grid(n_dst, DHID), 128, 0, stream>>>(
      agg, p.Ws, p.bs, t64a, n_dst, HC_C, DHID);
  const int lnb = (n_dst + 7) / 8;
  ln_residual_kernel<<<lnb, 256, 0, stream>>>(t64a, x_dst, p.g1, p.b1n, t64b,
                                              n_dst);
  gemm_bias_wmma<true><<<gemm_grid(n_dst, HC_C), 128, 0, stream>>>(
      t64b, p.W1, p.b1f, vbuf, n_dst, DHID, HC_C);  // relu fused
  gemm_bias_wmma<false><<<gemm_grid(n_dst, DHID), 128, 0, stream>>>(
      vbuf, p.W2, p.b2f, t64a, n_dst, HC_C, DHID);
  // out may alias x_dst: each thread reads its own elements before writing
  ln_residual_kernel<<<lnb, 256, 0, stream>>>(t64a, x_dst, p.g2, p.b2n, out,
                                              n_dst);
}

extern "C" void kernel_launch(void* const* d_in, const int* in_sizes, int n_in,
                              void* d_out, int out_size, void* d_ws,
                              size_t ws_size, hipStream_t stream) {
  // setup_inputs() insertion order, nested param dicts flattened in order
  const float* nodes  = (const float*)d_in[0];
  const float* actors = (const float*)d_in[1];
  const float* np_[16];
  const float* ap_[16];
  for (int i = 0; i < 16; ++i) np_[i] = (const float*)d_in[2 + i];
  for (int i = 0; i < 16; ++i) ap_[i] = (const float*)d_in[18 + i];
  const int* ne_src = (const int*)d_in[34];
  const int* ne_dst = (const int*)d_in[35];
  const int* na_src = (const int*)d_in[36];
  const int* na_dst = (const int*)d_in[37];

  // workspace layout (floats) ~ 97MB, L2-resident
  float* ws   = (float*)d_ws;
  float* xbuf = ws;                                  // 50000*64  node features
  float* abuf = xbuf + (size_t)N_NODES_C * DHID;     // 1024*64   actor features
  float* qbuf = abuf + (size_t)N_ACTORS_C * DHID;    // 50000*128 q / agg
  float* kbuf = qbuf + (size_t)N_NODES_C * HC_C;     // 50000*128 k / temps
  float* vbuf = kbuf + (size_t)N_NODES_C * HC_C;     // 50000*128 v / ff hidden
  float* exb  = vbuf + (size_t)N_NODES_C * HC_C;     // 800000*2  alpha/ex
  float* amax = exb  + (size_t)NE_EDGES_C * HEADS;   // 50000*2
  float* den  = amax + (size_t)N_NODES_C * HEADS;    // 50000*2

  hipMemcpyAsync(xbuf, nodes, (size_t)N_NODES_C * DHID * sizeof(float),
                 hipMemcpyDeviceToDevice, stream);

  // node self-attention blocks (in-place update of xbuf)
  for (int l = 0; l < 2; ++l) {
    LayerP p = layer_params(np_, l);
    run_block(xbuf, N_NODES_C, xbuf, N_NODES_C, ne_src, ne_dst, NE_EDGES_C, p,
              xbuf, qbuf, kbuf, vbuf, exb, amax, den, stream);
  }

  // node -> actor cross-attention blocks (in-place update of abuf)
  hipMemcpyAsync(abuf, actors, (size_t)N_ACTORS_C * DHID * sizeof(float),
                 hipMemcpyDeviceToDevice, stream);
  for (int l = 0; l < 2; ++l) {
    LayerP p = layer_params(ap_, l);
    run_block(xbuf, N_NODES_C, abuf, N_ACTORS_C, na_src, na_dst, NA_EDGES_C, p,
              abuf, qbuf, kbuf, vbuf, exb, amax, den, stream);
  }

  hipMemcpyAsync(d_out, abuf, (size_t)N_ACTORS_C * DHID * sizeof(float),
                 hipMemcpyDeviceToDevice, stream);
}